// LowPass_51178830299245
// MI455X (gfx1250) — compile-verified
//
#include <hip/hip_runtime.h>
#include <hip/hip_bf16.h>

// ---------------------------------------------------------------------------
// filtfilt (2nd-order Butterworth, PADLEN=9) for 1024 rows x 48000 f32.
//
// Per-row state-space block processing entirely in LDS:
//   * 16-sample blocks; zero-state response Y0 = H * X as 16x16x16 matmuls
//     via V_WMMA_F32_16X16X4_F32 (16 blocks per wave per issue).
//   * inter-block 2-state recurrence z' = M z + d via chunked scan
//     (constant M = A^16); inter-chunk combine is an 8-step Hillis-Steele
//     affine scan with precomputed Mc^(2^s).
//   * rank-2 output correction from zero-input responses k1, k2.
//   * backward pass = same routine templated with a reversed index map over
//     the full zero-padded physical buffer (110 leading zeros are harmless
//     through a causal zero-state filter); it also un-flips in place.
// No bounds predication anywhere in the hot path.
// ---------------------------------------------------------------------------

typedef float v2f __attribute__((ext_vector_type(2)));
typedef float v8f __attribute__((ext_vector_type(8)));

#define ROWS      1024
#define N_SAMP    48000
#define PAD       9
#define NP        (N_SAMP + 2*PAD)     // 48018 valid padded length
#define SUPER     188                  // superblocks of 16 blocks
#define NPHYS     (SUPER * 256)        // 48128 physical buffer (3008 blocks)
#define NBLK      (SUPER * 16)         // 3008 real blocks
#define NBPAD     3072                 // blocks covered by scan (256*12)
#define CHUNK     12
#define NTHREADS  256

// LDS layout (in floats)
#define OFF_BUF    0
#define OFF_DARR   (NPHYS)                   // 2*NBPAD floats (d, then z in-place)
#define OFF_EARR   (OFF_DARR + 2*NBPAD)      // 512
#define OFF_CARR   (OFF_EARR + 512)          // 512
#define OFF_RED    (OFF_CARR + 512)          // 256
#define OFF_CONST  (OFF_RED + 256)           // 128
#define SMEM_FLOATS (OFF_CONST + 128)
#define SMEM_BYTES  (SMEM_FLOATS * 4)

// const block: [0..15]=h  [16..31]=k1  [32..47]=k2
// [48]=b0 [49]=bb1 [50]=bb2 [51]=a1 [52]=a2  [53..56]=M (A^16)
// [61]=inv_scale [62]=scale  [64+4s .. 67+4s]=Mc^(2^s), s=0..7

__device__ __forceinline__ void mat2mul(const float* X, const float* Y, float* R) {
    R[0] = X[0]*Y[0] + X[1]*Y[2];
    R[1] = X[0]*Y[1] + X[1]*Y[3];
    R[2] = X[2]*Y[0] + X[3]*Y[2];
    R[3] = X[2]*Y[1] + X[3]*Y[3];
}

template <int REV>
__device__ void iir_pass(float* sm, const v2f (&aA)[4])
{
    float* buf  = sm + OFF_BUF;
    float* darr = sm + OFF_DARR;
    float* eArr = sm + OFF_EARR;
    float* cArr = sm + OFF_CARR;
    const float* K = sm + OFF_CONST;

    const float b0  = K[48], bb1 = K[49], bb2 = K[50], a1 = K[51], a2 = K[52];
    const float M00 = K[53], M01 = K[54], M10 = K[55], M11 = K[56];

    const int tid  = threadIdx.x;
    const int lane = tid & 31;
    const int wave = tid >> 5;        // 0..7
    const int j    = lane & 15;       // block column (N)
    const int half = lane >> 4;       // 0: K=4k+{0,1}, rows 0..7 ; 1: K=4k+{2,3}, rows 8..15

    // ---- Stage 1: zero-state block responses (Y0 = H * X) + per-block d ----
    for (int sb = wave; sb < SUPER; sb += NTHREADS/32) {
        const int pb = sb * 256 + j * 16;        // first sample of block (sb*16 + j)

        // gather all 8 B-operand values first, then issue 4 WMMAs back-to-back
        v2f bB[4];
#pragma unroll
        for (int k = 0; k < 4; ++k) {
            const int s0 = 4*k + 2*half;
            if (REV) {
                const int q = NPHYS - 1 - (pb + s0);
                bB[k].x = buf[q];
                bB[k].y = buf[q - 1];
            } else {
                bB[k].x = buf[pb + s0];
                bB[k].y = buf[pb + s0 + 1];
            }
        }

        v8f c = {0.f, 0.f, 0.f, 0.f, 0.f, 0.f, 0.f, 0.f};
#pragma unroll
        for (int k = 0; k < 4; ++k)
            c = __builtin_amdgcn_wmma_f32_16x16x4_f32(false, aA[k], false, bB[k],
                                                      (short)0, c, false, false);

        // write Y0 in place of X (lane holds samples r+8*half of block j)
        if (REV) {
            const int qb = NPHYS - 1 - (pb + 8*half);
#pragma unroll
            for (int r = 0; r < 8; ++r) buf[qb - r] = c[r];
        } else {
            float* yp = buf + pb + 8*half;
#pragma unroll
            for (int r = 0; r < 8; ++r) yp[r] = c[r];
        }

        // zero-state end state d (upper-half lanes own rows 14/15 = c[6],c[7];
        // their bB[3] holds logical samples 14,15 of the block)
        if (half) {
            const float x14 = bB[3].x, x15 = bB[3].y;
            const float y15 = c[7],    y14 = c[6];
            const float z1_14 = y15 - b0 * x15;
            const float z1_13 = y14 - b0 * x14;
            const float z2_14 = bb2 * x14 - a2 * z1_13;
            const int bk = sb * 16 + j;
            darr[2*bk    ] = bb1 * x15 - a1 * z1_14 + z2_14;
            darr[2*bk + 1] = bb2 * x15 - a2 * z1_14;
        }
    }
    // zero d for scan-padding blocks [NBLK, NBPAD)
    for (int i = 2*NBLK + tid; i < 2*NBPAD; i += NTHREADS) darr[i] = 0.f;
    __syncthreads();

    // ---- Stage 2: scan z_{k+1} = M z_k + d_k over blocks -------------------
    // chunk-local zero-state accumulation: e = sum M^(CHUNK-1-i) d_i
    {
        float e1 = 0.f, e2 = 0.f;
        const int base = tid * CHUNK;
#pragma unroll 4
        for (int i = 0; i < CHUNK; ++i) {
            const int bk = base + i;
            const float d1 = darr[2*bk], d2 = darr[2*bk + 1];
            const float t1 = M00*e1 + M01*e2 + d1;
            const float t2 = M10*e1 + M11*e2 + d2;
            e1 = t1; e2 = t2;
        }
        eArr[2*tid] = e1; eArr[2*tid + 1] = e2;
    }
    __syncthreads();

    // inter-chunk combine: Hillis-Steele affine scan, matrix per level is the
    // constant Mc^(2^s).  8 steps, ping-pong eArr <-> cArr.
    {
        float* src = eArr;
        float* dst = cArr;
#pragma unroll
        for (int s = 0; s < 8; ++s) {
            const float P0 = K[64+4*s], P1 = K[65+4*s];
            const float P2 = K[66+4*s], P3 = K[67+4*s];
            const int dlt = 1 << s;
            float f1 = src[2*tid], f2 = src[2*tid + 1];
            if (tid >= dlt) {
                const float g1 = src[2*(tid-dlt)], g2 = src[2*(tid-dlt) + 1];
                f1 += P0*g1 + P1*g2;
                f2 += P2*g1 + P3*g2;
            }
            dst[2*tid] = f1; dst[2*tid + 1] = f2;
            __syncthreads();
            float* t = src; src = dst; dst = t;
        }
        // carry into chunk t = inclusive scan at t-1
        float z1 = 0.f, z2 = 0.f;
        if (tid > 0) { z1 = src[2*(tid-1)]; z2 = src[2*(tid-1) + 1]; }

        // chunk replay: overwrite darr with the state ENTERING each block
        const int base = tid * CHUNK;
#pragma unroll 4
        for (int i = 0; i < CHUNK; ++i) {
            const int bk = base + i;
            const float d1 = darr[2*bk], d2 = darr[2*bk + 1];
            darr[2*bk] = z1; darr[2*bk + 1] = z2;
            const float t1 = M00*z1 + M01*z2 + d1;
            const float t2 = M10*z1 + M11*z2 + d2;
            z1 = t1; z2 = t2;
        }
    }
    __syncthreads();

    // ---- Stage 3: rank-2 output correction Y += k1*z1 + k2*z2 --------------
    for (int p = tid; p < NPHYS; p += NTHREADS) {
        const int bk = p >> 4, s = p & 15;
        const float zz1 = darr[2*bk], zz2 = darr[2*bk + 1];
        const float corr = K[16 + s] * zz1 + K[32 + s] * zz2;
        if (REV) buf[NPHYS - 1 - p] += corr;
        else     buf[p]             += corr;
    }
    __syncthreads();
}

__global__ __launch_bounds__(NTHREADS)
void filtfilt_kernel(const float* __restrict__ x,
                     const float* __restrict__ bcoef,
                     const float* __restrict__ acoef,
                     float* __restrict__ out)
{
    extern __shared__ float sm[];
    float* buf = sm + OFF_BUF;
    float* red = sm + OFF_RED;
    float* K   = sm + OFF_CONST;

    const int tid = threadIdx.x;
    const int row = blockIdx.x;
    const float4* x4 = reinterpret_cast<const float4*>(x + (size_t)row * N_SAMP);

    // load row into LDS (B128 global loads) + local max-abs
    float lm = 0.f;
    for (int i = tid; i < N_SAMP/4; i += NTHREADS) {
        const float4 v = x4[i];
        float* bp = buf + PAD + 4*i;
        bp[0] = v.x; bp[1] = v.y; bp[2] = v.z; bp[3] = v.w;
        lm = fmaxf(lm, fmaxf(fmaxf(fabsf(v.x), fabsf(v.y)),
                             fmaxf(fabsf(v.z), fabsf(v.w))));
    }
    red[tid] = lm;
    __syncthreads();
    for (int s = NTHREADS / 2; s > 0; s >>= 1) {
        if (tid < s) red[tid] = fmaxf(red[tid], red[tid + s]);
        __syncthreads();
    }

    // thread 0: scale + all filter constants
    if (tid == 0) {
        const float scale = red[0];
        K[62] = scale;
        K[61] = 1.f / scale;
        const float ia0 = 1.f / acoef[0];
        const float b0 = bcoef[0]*ia0, b1 = bcoef[1]*ia0, b2 = bcoef[2]*ia0;
        const float a1 = acoef[1]*ia0, a2 = acoef[2]*ia0;
        const float bb1 = b1 - a1*b0, bb2 = b2 - a2*b0;
        K[48]=b0; K[49]=bb1; K[50]=bb2; K[51]=a1; K[52]=a2;
        // impulse response h[0..15]
        float s1 = bb1, s2 = bb2;
        K[0] = b0;
        for (int t = 1; t < 16; ++t) {
            K[t] = s1;
            const float n1 = -a1*s1 + s2, n2 = -a2*s1;
            s1 = n1; s2 = n2;
        }
        // zero-input responses k1 (z_in=e1), k2 (z_in=e2)
        float p1 = 1.f, p2 = 0.f, q1 = 0.f, q2 = 1.f;
        for (int t = 0; t < 16; ++t) {
            K[16 + t] = p1; K[32 + t] = q1;
            const float np1 = -a1*p1 + p2, np2 = -a2*p1;
            const float nq1 = -a1*q1 + q2, nq2 = -a2*q1;
            p1 = np1; p2 = np2; q1 = nq1; q2 = nq2;
        }
        // M = A^16
        float A[4] = { -a1, 1.f, -a2, 0.f };
        float T[4], U[4];
        mat2mul(A, A, T); mat2mul(T, T, U); mat2mul(U, U, T); mat2mul(T, T, U);
        K[53]=U[0]; K[54]=U[1]; K[55]=U[2]; K[56]=U[3];
        // Mc = M^12, then powers Mc^(2^s) for the log-scan
        float M2[4], M4[4], M8[4], P[4], Q[4];
        mat2mul(U, U, M2); mat2mul(M2, M2, M4); mat2mul(M4, M4, M8);
        mat2mul(M8, M4, P);                                  // Mc = M^12
        for (int s = 0; s < 8; ++s) {
            K[64+4*s]=P[0]; K[65+4*s]=P[1]; K[66+4*s]=P[2]; K[67+4*s]=P[3];
            mat2mul(P, P, Q);
            P[0]=Q[0]; P[1]=Q[1]; P[2]=Q[2]; P[3]=Q[3];
        }
    }
    __syncthreads();

    // normalize
    const float inv = K[61];
    for (int i = tid; i < N_SAMP; i += NTHREADS) buf[PAD + i] *= inv;
    __syncthreads();

    // odd-reflection pads
    if (tid < PAD) {
        buf[tid] = 2.f * buf[PAD] - buf[PAD + 8 - tid];
    } else if (tid >= 16 && tid < 16 + PAD) {
        const int j2 = tid - 16;
        buf[PAD + N_SAMP + j2] = 2.f * buf[PAD + N_SAMP - 1] - buf[PAD + N_SAMP - 3 - j2];
    }
    // zero physical tail [NP, NPHYS)
    for (int i = NP + tid; i < NPHYS; i += NTHREADS) buf[i] = 0.f;
    __syncthreads();

    // Per-lane A-matrix operands: A[M][K] = h[M-K] (lower-triangular Toeplitz).
    // Same for both passes.
    const int lane = tid & 31;
    const int jj   = lane & 15;
    const int half = lane >> 4;
    v2f aA[4];
#pragma unroll
    for (int k = 0; k < 4; ++k) {
        const int k0 = 4*k + 2*half;
        aA[k].x = (jj >= k0    ) ? K[jj - k0    ] : 0.f;
        aA[k].y = (jj >= k0 + 1) ? K[jj - k0 - 1] : 0.f;
    }

    iir_pass<0>(sm, aA);   // forward

    // re-zero tail: reversed stream must start with true zeros
    for (int i = NP + tid; i < NPHYS; i += NTHREADS) buf[i] = 0.f;
    __syncthreads();

    iir_pass<1>(sm, aA);   // backward (reversed mapping; un-flips in place)

    // clip, un-normalize, trim pads, store (B128 global stores)
    const float scale = K[62];
    float4* o4 = reinterpret_cast<float4*>(out + (size_t)row * N_SAMP);
    for (int i = tid; i < N_SAMP/4; i += NTHREADS) {
        const float* bp = buf + PAD + 4*i;
        float4 v;
        v.x = fminf(fmaxf(bp[0], -1.f), 1.f) * scale;
        v.y = fminf(fmaxf(bp[1], -1.f), 1.f) * scale;
        v.z = fminf(fmaxf(bp[2], -1.f), 1.f) * scale;
        v.w = fminf(fmaxf(bp[3], -1.f), 1.f) * scale;
        o4[i] = v;
    }
}

extern "C" void kernel_launch(void* const* d_in, const int* in_sizes, int n_in,
                              void* d_out, int out_size, void* d_ws, size_t ws_size,
                              hipStream_t stream)
{
    (void)in_sizes; (void)n_in; (void)d_ws; (void)ws_size; (void)out_size;
    const float* x = (const float*)d_in[0];   // (16,64,48000) f32
    const float* b = (const float*)d_in[1];   // (3,) f32
    const float* a = (const float*)d_in[2];   // (3,) f32
    float* out = (float*)d_out;

    static_assert(SMEM_BYTES <= 320 * 1024, "LDS budget exceeded");
    (void)hipFuncSetAttribute(reinterpret_cast<const void*>(&filtfilt_kernel),
                              hipFuncAttributeMaxDynamicSharedMemorySize,
                              SMEM_BYTES);
    filtfilt_kernel<<<ROWS, NTHREADS, SMEM_BYTES, stream>>>(x, b, a, out);
}